// DCN_layer_2293512536710
// MI455X (gfx1250) — compile-verified
//
#include <hip/hip_runtime.h>
#include <hip/hip_bf16.h>
#include <math.h>

typedef __attribute__((ext_vector_type(2))) float v2f;
typedef __attribute__((ext_vector_type(8))) float v8f;

#define Bn   4
#define Cn   128
#define Hn   128
#define Wn   128
#define On   128
#define Kn   9
#define HWn  (Hn * Wn)          // 16384
#define CK   (Cn * Kn)          // 1152
#define LDA  292                // padded LDS row stride (floats), keeps 8B align, spreads banks
#define MT   32                 // M tile: 32 pixels per block (2 x 16-row WMMA sub-tiles)

static __device__ __forceinline__ int iclamp(int v, int lo, int hi) {
    return v < lo ? lo : (v > hi ? hi : v);
}

// ------------------- K0: input_feat NCHW -> NHWC (for coalesced channel gathers) -------------------
__global__ void nchw_to_nhwc(const float* __restrict__ in, float* __restrict__ out) {
    __shared__ float tile[32][33];
    int tb = blockIdx.x;
    int pt = tb % (HWn / 32); tb /= (HWn / 32);
    int ct = tb % (Cn / 32);  tb /= (Cn / 32);
    int b  = tb;
    int tx = threadIdx.x & 31;
    int ty = threadIdx.x >> 5;            // 0..7
    int c0 = ct * 32, p0 = pt * 32;
    const float* src = in + (size_t)b * Cn * HWn;
    #pragma unroll
    for (int r = 0; r < 32; r += 8)
        tile[ty + r][tx] = src[(size_t)(c0 + ty + r) * HWn + p0 + tx];
    __syncthreads();
    float* dst = out + (size_t)b * HWn * Cn;
    #pragma unroll
    for (int r = 0; r < 32; r += 8)
        dst[(size_t)(p0 + ty + r) * Cn + c0 + tx] = tile[tx][ty + r];
}

// ------------------- K1: offset conv (27ch) + sigmoid + bilinear-corner precompute -------------------
// Stores, per (b,k,pixel): mask-premultiplied bilinear weights (float4) and the four
// corner base offsets already scaled by C: (y*W+x)*C  (int4) -> hot kernel does base[id + c].
__global__ void offset_conv(const float* __restrict__ xin,
                            const float* __restrict__ inter,
                            const float* __restrict__ ow,   // [27][256][3][3]
                            const float* __restrict__ ob,   // [27]
                            float4* __restrict__ sampW,
                            int4*  __restrict__ sampI) {
    int t   = blockIdx.x * blockDim.x + threadIdx.x;
    int b   = t / HWn;
    int pix = t % HWn;
    int y = pix >> 7, x = pix & (Wn - 1);

    float acc[27];
    #pragma unroll
    for (int o = 0; o < 27; ++o) acc[o] = ob[o];   // thread-uniform -> s_load

    for (int ky = 0; ky < 3; ++ky) {
        int yy = y + ky - 1;
        if (yy < 0 || yy >= Hn) continue;          // zero padding
        for (int kx = 0; kx < 3; ++kx) {
            int xx = x + kx - 1;
            if (xx < 0 || xx >= Wn) continue;
            const float* f0 = xin   + (size_t)b * Cn * HWn + yy * Wn + xx;
            const float* f1 = inter + (size_t)b * Cn * HWn + yy * Wn + xx;
            const float* wp = ow + ky * 3 + kx;
            for (int c = 0; c < Cn; ++c) {         // weights thread-uniform -> scalar loads
                float v = f0[(size_t)c * HWn];     // coalesced across lanes (x-adjacent)
                #pragma unroll
                for (int o = 0; o < 27; ++o)
                    acc[o] = fmaf(v, wp[(o * 256 + c) * 9], acc[o]);
            }
            for (int c = 0; c < Cn; ++c) {
                float v = f1[(size_t)c * HWn];
                #pragma unroll
                for (int o = 0; o < 27; ++o)
                    acc[o] = fmaf(v, wp[(o * 256 + 128 + c) * 9], acc[o]);
            }
        }
    }

    #pragma unroll
    for (int k = 0; k < 9; ++k) {
        float dy = acc[2 * k], dx = acc[2 * k + 1];
        float m  = 1.0f / (1.0f + __expf(-acc[18 + k]));   // sigmoid(mask)
        int ky = k / 3, kx = k - ky * 3;
        float py = dy + (float)(y - 1 + ky);
        float px = dx + (float)(x - 1 + kx);
        float y0f = floorf(py), x0f = floorf(px);
        float wy = py - y0f, wx = px - x0f;
        int y0 = (int)y0f, x0 = (int)x0f;
        int y1 = y0 + 1,   x1 = x0 + 1;
        float vy0 = (y0 >= 0 && y0 < Hn) ? 1.0f : 0.0f;
        float vy1 = (y1 >= 0 && y1 < Hn) ? 1.0f : 0.0f;
        float vx0 = (x0 >= 0 && x0 < Wn) ? 1.0f : 0.0f;
        float vx1 = (x1 >= 0 && x1 < Wn) ? 1.0f : 0.0f;
        float4 w;
        w.x = (1.0f - wy) * (1.0f - wx) * vy0 * vx0 * m;
        w.y = (1.0f - wy) * wx          * vy0 * vx1 * m;
        w.z = wy          * (1.0f - wx) * vy1 * vx0 * m;
        w.w = wy          * wx          * vy1 * vx1 * m;
        int y0c = iclamp(y0, 0, Hn - 1);
        int x0c = iclamp(x0, 0, Wn - 1);
        int y1c = iclamp(y1, 0, Hn - 1);
        int x1c = iclamp(x1, 0, Wn - 1);
        int4 id;                                   // element offsets into NHWC image, pre-scaled by C
        id.x = ((y0c << 7) + x0c) << 7;            // (y0*W + x0) * C
        id.y = ((y0c << 7) + x1c) << 7;
        id.z = ((y1c << 7) + x0c) << 7;
        id.w = ((y1c << 7) + x1c) << 7;
        size_t si = (size_t)(b * 9 + k) * HWn + pix;
        sampW[si] = w;
        sampI[si] = id;
    }
}

// ------------------- K2: fused bilinear-sample + mask + WMMA f32 GEMM -------------------
// Block: 256 thr (8 waves). M-tile = 32 pixels (2 WMMA sub-tiles sharing each B fragment),
// each wave owns 16 output channels. K = 1152 as 4 chunks of 288 (32 ch x 9 taps).
__global__ void __launch_bounds__(256, 2)
dcn_gemm(const float* __restrict__ inNHWC,
         const float4* __restrict__ sampW,
         const int4*  __restrict__ sampI,
         const float* __restrict__ weight,   // [128][1152] == [O][C*9]
         const float* __restrict__ bias,
         float* __restrict__ out) {
    __shared__ float Asm[MT * LDA];          // A tile: [32 pix][288 k-slice]
    __shared__ float Osm[MT * 128];          // output transpose staging

    const int tile = blockIdx.x;
    const int b    = tile >> 9;              // / (HW/32)
    const int pix0 = (tile & 511) << 5;      // 32 consecutive x within a row (W=128 | 32)

    const int t    = threadIdx.x;
    const int lane = t & 31;
    const int wv   = t >> 5;                 // wave id 0..7
    const int n0   = wv << 4;                // this wave's output-channel base
    const int nidx = lane & 15;              // N column (B/D fragments)
    const int h    = lane >> 4;              // K-half select

    float bv = bias[n0 + nidx];
    v8f acc0, acc1;
    #pragma unroll
    for (int r = 0; r < 8; ++r) { acc0[r] = bv; acc1[r] = bv; }

    const int cl  = t & 31;                  // channel-local within chunk (coalesced gathers)
    const int grp = t >> 5;                  // pixel-quad group, uniform within wave
    const float* base = inNHWC + (size_t)b * HWn * Cn;   // scalar base -> SGPR+voffset loads

    for (int chunk = 0; chunk < 4; ++chunk) {
        __syncthreads();
        // ---- build A tile: cols[pix][c*9+k] for c in [chunk*32, chunk*32+32) ----
        // Unroll capped (VGPRs < 256); indices unsigned so corner addresses zero-extend
        // (folds into SADDR+voffset form) instead of per-corner sext chains.
        const unsigned uc = (unsigned)((chunk << 5) + cl);
        #pragma unroll 1
        for (int pp4 = 0; pp4 < 4; ++pp4) {  // 4 pixels per group, 9 taps each (no div/mod)
            const int pp = (grp << 2) + pp4;
            const size_t sbase = (size_t)b * 9 * HWn + pix0 + pp;
            #pragma unroll 3
            for (int k = 0; k < 9; ++k) {
                size_t si = sbase + (size_t)k * HWn;     // wave-uniform
                int4   id = sampI[si];
                float4 wg = sampW[si];
                float v = wg.x * base[(unsigned)id.x + uc]
                        + wg.y * base[(unsigned)id.y + uc]
                        + wg.z * base[(unsigned)id.z + uc]
                        + wg.w * base[(unsigned)id.w + uc];
                Asm[pp * LDA + cl * 9 + k] = v;
            }
        }
        __syncthreads();

        // ---- 72 steps x (1 B-fragment, 2 A-fragments, 2 WMMA) over this 288-wide K slice ----
        const float* wrow = weight + (size_t)(n0 + nidx) * CK + chunk * 288;
        __builtin_prefetch(wrow + 288, 0, 1);   // global_prefetch_b8: next chunk's weights
        const float* arow0 = &Asm[(lane & 15) * LDA];
        const float* arow1 = arow0 + 16 * LDA;
        #pragma unroll 4
        for (int s = 0; s < 72; ++s) {
            int kk = (s << 2) + (h << 1);
            v2f w  = *(const v2f*)(wrow  + kk);  // global_load_b64, L2-hot, shared by 2 WMMA
            v2f a0 = *(const v2f*)(arow0 + kk);  // ds_load b64, 8B aligned
            v2f a1 = *(const v2f*)(arow1 + kk);
            acc0 = __builtin_amdgcn_wmma_f32_16x16x4_f32(
                false, a0, false, w, (short)0, acc0, false, false);
            acc1 = __builtin_amdgcn_wmma_f32_16x16x4_f32(
                false, a1, false, w, (short)0, acc1, false, false);
        }
    }

    // ---- transpose through LDS for coalesced NCHW stores ----
    __syncthreads();
    #pragma unroll
    for (int r = 0; r < 8; ++r) {
        int pp = r + (h << 3);               // D layout: VGPR r -> row r (+8 for upper half)
        Osm[pp * 128 + n0 + nidx]        = acc0[r];
        Osm[(pp + 16) * 128 + n0 + nidx] = acc1[r];
    }
    __syncthreads();
    {
        int chb = t >> 5;                    // 0..7
        int pl  = t & 31;                    // pixel within tile (lane-contiguous, 128B runs)
        float* obase = out + (size_t)b * On * HWn + pix0 + pl;
        #pragma unroll
        for (int cc = 0; cc < 16; ++cc) {
            int ch = cc * 8 + chb;
            obase[(size_t)ch * HWn] = Osm[pl * 128 + ch];
        }
    }
}

extern "C" void kernel_launch(void* const* d_in, const int* in_sizes, int n_in,
                              void* d_out, int out_size, void* d_ws, size_t ws_size,
                              hipStream_t stream) {
    const float* input_feat = (const float*)d_in[0];
    const float* inter      = (const float*)d_in[1];
    const float* offset_w   = (const float*)d_in[2];
    const float* offset_b   = (const float*)d_in[3];
    const float* weight     = (const float*)d_in[4];
    const float* bias       = (const float*)d_in[5];
    float* out = (float*)d_out;

    char* ws = (char*)d_ws;
    size_t nhwcBytes = (size_t)Bn * HWn * Cn * sizeof(float);        // 32 MB
    size_t sampBytes = (size_t)Bn * Kn * HWn * sizeof(float4);       // 9.4 MB each
    float*  inNHWC = (float*)ws;
    float4* sampW  = (float4*)(ws + nhwcBytes);
    int4*   sampI  = (int4*)(ws + nhwcBytes + sampBytes);

    nchw_to_nhwc<<<Bn * (Cn / 32) * (HWn / 32), 256, 0, stream>>>(input_feat, inNHWC);
    offset_conv<<<(Bn * HWn) / 256, 256, 0, stream>>>(input_feat, inter, offset_w, offset_b,
                                                      sampW, sampI);
    dcn_gemm<<<(Bn * HWn) / MT, 256, 0, stream>>>(inNHWC, sampW, sampI, weight, bias, out);
}